// HybridEstimator_65481071406759
// MI455X (gfx1250) — compile-verified
//
#include <hip/hip_runtime.h>

// cos(x) over 262144 x 128 f32 = 33.55M elements. Pure HBM stream:
// 256 MB total traffic -> ~11us roofline at 23.3 TB/s. Strategy:
//  - b128 vector loads/stores, non-temporal (stream > 192MB L2, zero reuse)
//  - hardware v_cos_f32 (input pre-scaled by 1/(2*pi) inline constant:
//    AMD V_COS_F32 takes revolutions)
//  - 8 elements/thread: two b128 loads in flight per thread for MLP
//  - NO device-side branching: host launches exactly out_size/8 threads;
//    remainder (never hit for this shape) handled by a separate tail kernel.

typedef float v4f __attribute__((ext_vector_type(4)));

#define INV_2PI 0.15915494309189535f

__global__ __launch_bounds__(256) void cos_stream_kernel(const float* __restrict__ in,
                                                         float* __restrict__ out) {
    // Each thread handles two consecutive float4 groups (8 floats). Straight-line.
    int t = blockIdx.x * blockDim.x + threadIdx.x;
    long long i0 = (long long)t * 2;

    const v4f* __restrict__ vin  = (const v4f*)in;
    v4f* __restrict__ vout = (v4f*)out;

    // Two 128-bit non-temporal loads issued back-to-back (2 loads in flight).
    v4f a = __builtin_nontemporal_load(vin + i0);
    v4f b = __builtin_nontemporal_load(vin + i0 + 1);

    v4f ra, rb;
    ra.x = __builtin_amdgcn_cosf(a.x * INV_2PI);
    ra.y = __builtin_amdgcn_cosf(a.y * INV_2PI);
    ra.z = __builtin_amdgcn_cosf(a.z * INV_2PI);
    ra.w = __builtin_amdgcn_cosf(a.w * INV_2PI);
    rb.x = __builtin_amdgcn_cosf(b.x * INV_2PI);
    rb.y = __builtin_amdgcn_cosf(b.y * INV_2PI);
    rb.z = __builtin_amdgcn_cosf(b.z * INV_2PI);
    rb.w = __builtin_amdgcn_cosf(b.w * INV_2PI);

    __builtin_nontemporal_store(ra, vout + i0);
    __builtin_nontemporal_store(rb, vout + i0 + 1);
}

// Scalar tail (only launched if out_size % 2048 != 0 — never for this shape).
__global__ void cos_tail_kernel(const float* __restrict__ in,
                                float* __restrict__ out,
                                int base, int n) {
    int i = base + blockIdx.x * blockDim.x + threadIdx.x;
    if (i < n) {
        out[i] = __builtin_amdgcn_cosf(in[i] * INV_2PI);
    }
}

extern "C" void kernel_launch(void* const* d_in, const int* in_sizes, int n_in,
                              void* d_out, int out_size, void* d_ws, size_t ws_size,
                              hipStream_t stream) {
    const float* in = (const float*)d_in[0];
    float* out = (float*)d_out;

    // out_size = 262144 * 128 = 33,554,432 floats; divisible by 8.
    int main_elems = out_size & ~7;        // multiple of 8
    int threads = main_elems / 8;          // 8 floats per thread
    if (threads > 0) {
        int block = 256;
        int grid = threads / block;
        if (grid * block < threads) {
            // Fall back: shrink to exact multiple, push rest to tail.
            main_elems = (grid * block) * 8;
            threads = grid * block;
        }
        if (grid > 0) {
            cos_stream_kernel<<<grid, 256, 0, stream>>>(in, out);
        } else {
            main_elems = 0;
        }
    }
    int rem = out_size - main_elems;
    if (rem > 0) {
        int block = 256;
        int grid = (rem + block - 1) / block;
        cos_tail_kernel<<<grid, block, 0, stream>>>(in, out, main_elems, out_size);
    }
}